// VariationalClassifier_65481071399525
// MI455X (gfx1250) — compile-verified
//
#include <hip/hip_runtime.h>
#include <math.h>

// ---------------------------------------------------------------------------
// 16-qubit variational circuit simulator, fully fused, one WG per batch elem.
//   state part (re or im) lives entirely in LDS (256KB); gates fused into
//   16x16 real matrices applied with V_WMMA_F32_16X16X4_F32 (4 K-slices).
// Qubit q <-> flat bit (15-q). Group g covers qubits 4g..4g+3 <-> hex digit
// d = 3-g (bits 4d+3..4d), qubit 4g being the digit MSB.
// ---------------------------------------------------------------------------

#define NQ      16
#define DIM     65536
#define DEPTH   8
#define BATCH   128
#define BLOCK   1024
#define NWAVES  (BLOCK / 32)

typedef float v2f __attribute__((ext_vector_type(2)));
typedef float v8f __attribute__((ext_vector_type(8)));

// LDS layout (floats): S[65536] | MAT[8*4*256] | CMAG[64] | CS[32] | RED[512]
#define OFF_MAT   65536
#define OFF_CMAG  (OFF_MAT + 8192)
#define OFF_CS    (OFF_CMAG + 64)
#define OFF_RED   (OFF_CS + 32)
#define LDS_FLOATS (OFF_RED + NWAVES * 16)

__global__ __launch_bounds__(BLOCK) void
vqc_sim_kernel(const float* __restrict__ x, const float* __restrict__ theta,
               float* __restrict__ out) {
    extern __shared__ float smem[];
    float* S    = smem;
    float* MAT  = smem + OFF_MAT;   // [layer][group][row][col] row-major
    float* CMAG = smem + OFF_CMAG;  // [group][digit] magnitude factors
    float* CS   = smem + OFF_CS;    // c[16], s[16] of encoding angles
    float* RED  = smem + OFF_RED;

    const int tid  = threadIdx.x;
    const int b    = blockIdx.x;
    const int lane = tid & 31;
    const int wave = tid >> 5;
    const int n    = lane & 15;      // WMMA column / A row (M)
    const int half = lane >> 4;      // lane half selects K pair
    const int k0   = half << 1;      // K offset 0 or 2 within a K=4 slice

    // ---- encoding angle tables -------------------------------------------
    if (tid < NQ) {
        float h = 0.5f * x[b * NQ + tid];
        CS[tid]      = cosf(h);
        CS[16 + tid] = sinf(h);
    }
    __syncthreads();
    if (tid < 64) {
        int g = tid >> 4, dv = tid & 15;
        float m = 1.0f;
        #pragma unroll
        for (int k = 0; k < 4; ++k) {
            int bit = (dv >> (3 - k)) & 1;
            m *= bit ? CS[16 + 4 * g + k] : CS[4 * g + k];
        }
        CMAG[tid] = m;
    }

    // ---- fused layer matrices: (intra-group CZ) * kron(RY x4) ------------
    for (int e = tid; e < 8192; e += BLOCK) {
        int l = e >> 10, rem = e & 1023;
        int g = rem >> 8, i = (rem >> 4) & 15, j = rem & 15;
        float prod = 1.0f;
        #pragma unroll
        for (int k = 0; k < 4; ++k) {
            int q  = 4 * g + k;
            float h  = 0.5f * theta[l * NQ + q];
            float cc = cosf(h), ss = sinf(h);
            int rb = (i >> (3 - k)) & 1, cb = (j >> (3 - k)) & 1;
            prod *= (rb == cb) ? cc : (rb ? ss : -ss);   // RY = [[c,-s],[s,c]]
        }
        int czp = ((i >> 3) & (i >> 2)) ^ ((i >> 2) & (i >> 1)) ^ ((i >> 1) & i);
        MAT[e] = (czp & 1) ? -prod : prod;
    }
    __syncthreads();

    float acc[16];
    #pragma unroll
    for (int j = 0; j < 16; ++j) acc[j] = 0.0f;

    // ---- evolve real part, then imaginary part ---------------------------
    for (int part = 0; part < 2; ++part) {
        // fill: amp(i) = Mag(i) * (-i)^popcount(i); take re or im component
        for (int i = tid; i < DIM; i += BLOCK) {
            float mag = CMAG[(i >> 12) & 15] * CMAG[16 + ((i >> 8) & 15)] *
                        CMAG[32 + ((i >> 4) & 15)] * CMAG[48 + (i & 15)];
            int pc = __popc(i) & 3;
            float v;
            if (part == 0) v = (pc == 0) ? mag : ((pc == 2) ? -mag : 0.0f);
            else           v = (pc == 1) ? -mag : ((pc == 3) ? mag : 0.0f);
            S[i] = v;
        }
        __syncthreads();

        for (int l = 0; l < DEPTH; ++l) {
            // four group matmuls (axes commute; barrier between in-place passes)
            #pragma unroll
            for (int g = 0; g < 4; ++g) {
                const int sh = 4 * (3 - g);
                const int lowmask = (1 << sh) - 1;
                const float* Ur = MAT + (((l << 2) + g) << 8) + (n << 4);
                // A operand (gate matrix) is invariant across column blocks:
                // load once into registers so S[] stores can't force reloads.
                v2f a0, a1, a2, a3;
                a0.x = Ur[k0 + 0];  a0.y = Ur[k0 + 1];
                a1.x = Ur[k0 + 4];  a1.y = Ur[k0 + 5];
                a2.x = Ur[k0 + 8];  a2.y = Ur[k0 + 9];
                a3.x = Ur[k0 + 12]; a3.y = Ur[k0 + 13];
                for (int blk = wave; blk < 256; blk += NWAVES) {
                    const int col  = (blk << 4) + n;        // rest-index column
                    const int base = ((col & ~lowmask) << 4) | (col & lowmask);
                    v2f b0, b1, b2, b3;
                    b0.x = S[base + ((k0 + 0)  << sh)]; b0.y = S[base + ((k0 + 1)  << sh)];
                    b1.x = S[base + ((k0 + 4)  << sh)]; b1.y = S[base + ((k0 + 5)  << sh)];
                    b2.x = S[base + ((k0 + 8)  << sh)]; b2.y = S[base + ((k0 + 9)  << sh)];
                    b3.x = S[base + ((k0 + 12) << sh)]; b3.y = S[base + ((k0 + 13) << sh)];
                    v8f c = {0.f, 0.f, 0.f, 0.f, 0.f, 0.f, 0.f, 0.f};
                    c = __builtin_amdgcn_wmma_f32_16x16x4_f32(false, a0, false, b0, (short)0, c, false, false);
                    c = __builtin_amdgcn_wmma_f32_16x16x4_f32(false, a1, false, b1, (short)0, c, false, false);
                    c = __builtin_amdgcn_wmma_f32_16x16x4_f32(false, a2, false, b2, (short)0, c, false, false);
                    c = __builtin_amdgcn_wmma_f32_16x16x4_f32(false, a3, false, b3, (short)0, c, false, false);
                    #pragma unroll
                    for (int p = 0; p < 8; ++p)
                        S[base + ((p + (half << 3)) << sh)] = c[p];
                }
                __syncthreads();
            }
            // cross-group CZ: qubit pairs (3,4),(7,8),(11,12) -> bit pairs.
            // The final layer's diagonal +-1 phases cannot affect |amp|^2,
            // so skip the pass on the last layer.
            if (l != DEPTH - 1) {
                for (int i = tid; i < DIM; i += BLOCK) {
                    int par = ((i >> 12) & (i >> 11)) ^ ((i >> 8) & (i >> 7)) ^
                              ((i >> 4) & (i >> 3));
                    if (par & 1) S[i] = -S[i];
                }
                __syncthreads();
            }
        }

        // accumulate Z contributions of this part: out[b][j] += sign_j * v^2
        for (int i = tid; i < DIM; i += BLOCK) {
            float v = S[i], p = v * v;
            #pragma unroll
            for (int j = 0; j < 16; ++j)
                acc[j] += ((i >> j) & 1) ? -p : p;
        }
        __syncthreads();
    }

    // ---- reduce across the workgroup -------------------------------------
    #pragma unroll
    for (int j = 0; j < 16; ++j) {
        float v = acc[j];
        for (int o = 16; o >= 1; o >>= 1) v += __shfl_xor(v, o, 32);
        if (lane == 0) RED[wave * 16 + j] = v;
    }
    __syncthreads();
    if (tid < 16) {
        float s = 0.0f;
        #pragma unroll
        for (int w = 0; w < NWAVES; ++w) s += RED[w * 16 + tid];
        out[b * NQ + tid] = s;
    }
}

extern "C" void kernel_launch(void* const* d_in, const int* in_sizes, int n_in,
                              void* d_out, int out_size, void* d_ws, size_t ws_size,
                              hipStream_t stream) {
    const float* x     = (const float*)d_in[0];
    const float* theta = (const float*)d_in[1];
    float* out         = (float*)d_out;
    (void)in_sizes; (void)n_in; (void)out_size; (void)d_ws; (void)ws_size;

    size_t shmem = (size_t)LDS_FLOATS * sizeof(float);  // ~290 KB <= 320 KB/WGP
    vqc_sim_kernel<<<BATCH, BLOCK, shmem, stream>>>(x, theta, out);
}